// FilterLayer_39049842655925
// MI455X (gfx1250) — compile-verified
//
#include <hip/hip_runtime.h>
#include <math.h>

// FilterLayer: y = LayerNorm( mix * irfft(w .* rfft(x)) + (1-mix) * x )
// Realized as two f32 WMMA GEMM stages with on-the-fly DFT matrix generation.
// B=1024, S=128, H=256, all fp32.

typedef float v2f __attribute__((ext_vector_type(2)));
typedef float v8f __attribute__((ext_vector_type(8)));

#define SEQ 128
#define HID 256
#define NF  65          // rfft bins
#define KP  68          // padded freq rows (multiple of 4)
#define LDST 76         // LDS row stride (floats): bank-conflict-free for both phases
#define HCHUNK 64       // h columns per chunk (4 chunks)

__global__ __launch_bounds__(256)
void filter_layer_kernel(const float* __restrict__ x,
                         const float* __restrict__ cw,     // [65,256,2]
                         const float* __restrict__ fw,     // scalar
                         const float* __restrict__ gamma,  // [256]
                         const float* __restrict__ beta,   // [256]
                         float* __restrict__ out)
{
    __shared__ float sPr[KP * LDST];
    __shared__ float sPi[KP * LDST];

    const int b     = blockIdx.x;
    const int tid   = threadIdx.x;
    const int lane  = tid & 31;
    const int wid   = tid >> 5;        // 8 waves
    const int m16   = lane & 15;
    const int hihalf = lane >> 4;      // 0 or 1
    const int khalf = hihalf << 1;     // k offset 0 or 2 (A/B fragment layout)

    const float* xb = x  + (size_t)b * SEQ * HID;
    float*       ob = out + (size_t)b * SEQ * HID;

    const float TPO = 6.283185307179586f / 128.0f;  // 2*pi/128

    const float fwv  = fw[0];
    const float mixv = 1.0f / (1.0f + __expf(-fwv));
    const float onem = 1.0f - mixv;

    float sum[8], sq[8];
#pragma unroll
    for (int j = 0; j < 8; ++j) { sum[j] = 0.0f; sq[j] = 0.0f; }

    const int srow = wid * 16 + m16;   // stage-3 A row (output s) for this wave

    for (int hc = 0; hc < 4; ++hc) {
        // ---------------- Stage 1: P = W .* DFT(x) for h in [hc*64, hc*64+64) ----
        for (int tile = wid; tile < 20; tile += 8) {     // 5 f-tiles x 4 h-tiles
            const int mtile = tile >> 2;                 // 0..4
            const int nl    = tile & 3;                  // 0..3
            const int f  = mtile * 16 + m16;             // A row (frequency)
            const int hl = nl * 16 + m16;                // local h col
            const int h  = hc * HCHUNK + hl;             // global h col

            // cos/sin(2*pi*f*t/128) for t = khalf, khalf+1; rotate by t+=4 per k-step
            float c0, s0, c1, s1, cd, sd;
            {
                float a0 = (float)((f * (khalf + 0)) & 127) * TPO;
                float a1 = (float)((f * (khalf + 1)) & 127) * TPO;
                float ad = (float)((f * 4) & 127) * TPO;
                __sincosf(a0, &s0, &c0);
                __sincosf(a1, &s1, &c1);
                __sincosf(ad, &sd, &cd);
            }

            v8f accR = {}; v8f accI = {};
            const float* px = xb + (size_t)khalf * HID + h;
#pragma unroll 4
            for (int ks = 0; ks < 32; ++ks) {
                v2f bf; bf.x = px[0]; bf.y = px[HID];
                v2f ac; ac.x = c0;  ac.y = c1;
                v2f as_; as_.x = -s0; as_.y = -s1;
                accR = __builtin_amdgcn_wmma_f32_16x16x4_f32(
                           false, ac,  false, bf, (short)0, accR, false, false);
                accI = __builtin_amdgcn_wmma_f32_16x16x4_f32(
                           false, as_, false, bf, (short)0, accI, false, false);
                float nc0 = c0*cd - s0*sd, ns0 = s0*cd + c0*sd;
                float nc1 = c1*cd - s1*sd, ns1 = s1*cd + c1*sd;
                c0 = nc0; s0 = ns0; c1 = nc1; s1 = ns1;
                px += 4 * HID;
            }

            // epilogue: apply complex weight, store Pr/Pi (rows 65..67 zeroed)
#pragma unroll
            for (int j = 0; j < 8; ++j) {
                int fr = mtile * 16 + j + hihalf * 8;
                float pr = 0.0f, pi = 0.0f;
                if (fr < NF) {
                    float wr = cw[((size_t)fr * HID + h) * 2 + 0];
                    float wi = cw[((size_t)fr * HID + h) * 2 + 1];
                    float xr = accR[j], xi = accI[j];
                    pr = wr * xr - wi * xi;
                    pi = wr * xi + wi * xr;
                }
                if (fr < KP) {
                    sPr[fr * LDST + hl] = pr;
                    sPi[fr * LDST + hl] = pi;
                }
            }
        }
        __syncthreads();

        // ---------------- Stage 3: y = Dr*Pr + Di*Pi, mix with x, stash hidden ---
        for (int nl = 0; nl < 4; ++nl) {
            const int hl = nl * 16 + m16;
            const int h  = hc * HCHUNK + hl;

            float c0, s0, c1, s1, cd, sd;
            {
                float a0 = (float)((srow * (khalf + 0)) & 127) * TPO;
                float a1 = (float)((srow * (khalf + 1)) & 127) * TPO;
                float ad = (float)((srow * 4) & 127) * TPO;
                __sincosf(a0, &s0, &c0);
                __sincosf(a1, &s1, &c1);
                __sincosf(ad, &sd, &cd);
            }

            v8f acc = {};
#pragma unroll 4
            for (int ks = 0; ks < 17; ++ks) {
                const int f0 = ks * 4 + khalf;
                const float sc0 = (f0 == 0 || f0 == 64) ? (1.0f/128.0f) : (2.0f/128.0f);
                const float sc1 = 2.0f / 128.0f;
                v2f ar; ar.x =  sc0 * c0; ar.y =  sc1 * c1;
                v2f ai; ai.x = -sc0 * s0; ai.y = -sc1 * s1;
                v2f br; br.x = sPr[f0 * LDST + hl]; br.y = sPr[(f0 + 1) * LDST + hl];
                v2f bi; bi.x = sPi[f0 * LDST + hl]; bi.y = sPi[(f0 + 1) * LDST + hl];
                acc = __builtin_amdgcn_wmma_f32_16x16x4_f32(
                          false, ar, false, br, (short)0, acc, false, false);
                acc = __builtin_amdgcn_wmma_f32_16x16x4_f32(
                          false, ai, false, bi, (short)0, acc, false, false);
                float nc0 = c0*cd - s0*sd, ns0 = s0*cd + c0*sd;
                float nc1 = c1*cd - s1*sd, ns1 = s1*cd + c1*sd;
                c0 = nc0; s0 = ns0; c1 = nc1; s1 = ns1;
            }

#pragma unroll
            for (int j = 0; j < 8; ++j) {
                int sj = wid * 16 + j + hihalf * 8;
                float hv = mixv * acc[j] + onem * xb[(size_t)sj * HID + h];
                ob[(size_t)sj * HID + h] = hv;     // stash hidden
                sum[j] += hv;
                sq[j]  += hv * hv;
            }
        }
        __syncthreads();   // protect sPr/sPi before next chunk overwrites
    }

    // ---------------- LayerNorm over H for this wave's 16 rows --------------------
#pragma unroll
    for (int j = 0; j < 8; ++j) {
#pragma unroll
        for (int m = 1; m < 16; m <<= 1) {    // stays within each 16-lane half
            sum[j] += __shfl_xor(sum[j], m, 32);
            sq[j]  += __shfl_xor(sq[j],  m, 32);
        }
    }
    float mu[8], rs[8];
#pragma unroll
    for (int j = 0; j < 8; ++j) {
        mu[j] = sum[j] * (1.0f / 256.0f);
        float var = sq[j] * (1.0f / 256.0f) - mu[j] * mu[j];
        rs[j] = rsqrtf(var + 1e-12f);
    }

    for (int nt = 0; nt < 16; ++nt) {
        const int h = nt * 16 + m16;
        const float g  = gamma[h];
        const float bb = beta[h];
#pragma unroll
        for (int j = 0; j < 8; ++j) {
            int sj = wid * 16 + j + hihalf * 8;
            size_t idx = (size_t)sj * HID + h;
            float hv = ob[idx];                 // same-wave store->load: ordered
            ob[idx] = (hv - mu[j]) * rs[j] * g + bb;
        }
    }
}

extern "C" void kernel_launch(void* const* d_in, const int* in_sizes, int n_in,
                              void* d_out, int out_size, void* d_ws, size_t ws_size,
                              hipStream_t stream) {
    const float* x  = (const float*)d_in[0];
    const float* cw = (const float*)d_in[1];
    const float* fw = (const float*)d_in[2];
    const float* g  = (const float*)d_in[3];
    const float* be = (const float*)d_in[4];
    float* out = (float*)d_out;

    const int B = in_sizes[0] / (SEQ * HID);
    hipLaunchKernelGGL(filter_layer_kernel, dim3(B), dim3(256), 0, stream,
                       x, cw, fw, g, be, out);
}